// GCNSI_54692113547360
// MI455X (gfx1250) — compile-verified
//
#include <hip/hip_runtime.h>
#include <hip/hip_bf16.h>

typedef __attribute__((ext_vector_type(2))) float v2f;
typedef __attribute__((ext_vector_type(8))) float v8f;

#define NFEAT 4
#define HID   128

// ---------------- degree / normalization ----------------
__global__ void k_deg_init(float* __restrict__ deg, int n) {
  int i = blockIdx.x * blockDim.x + threadIdx.x;
  if (i < n) deg[i] = 1.0f;                 // self-loop contributes 1
}

__global__ void k_deg_count(const int* __restrict__ dst, float* __restrict__ deg, int e) {
  int i = blockIdx.x * blockDim.x + threadIdx.x;
  if (i < e) atomicAdd(&deg[dst[i]], 1.0f);
}

__global__ void k_dinv(float* __restrict__ deg, int n) {
  int i = blockIdx.x * blockDim.x + threadIdx.x;
  if (i < n) deg[i] = rsqrtf(deg[i]);       // deg >= 1 always (self loops)
}

// ---------------- dense GEMM via V_WMMA_F32_16X16X4_F32 ----------------
// C[16 x 128] tile per block: 8 waves, wave w owns cols [16w, 16w+16).
// A fragment (16x4 f32): lanes 0-15 hold K={k0,k0+1}, lanes 16-31 hold K={k0+2,k0+3}.
// B fragment (4x16 f32): same K split per half-wave, lane%16 = column.
// C/D (16x16 f32): VGPR j -> row j (lanes 0-15) / row j+8 (lanes 16-31), lane%16 = col.
template <int K>
__global__ __launch_bounds__(256) void k_gemm_wmma(const float* __restrict__ A,
                                                   const float* __restrict__ W,
                                                   float* __restrict__ C) {
  __shared__ float sA[16 * (K + 4)];        // padded stride: conflict-free LDS reads
  const int tid = threadIdx.x;
  const size_t row0 = (size_t)blockIdx.x * 16;

  for (int i = tid; i < 16 * K; i += 256) {
    int rr = i / K, cc = i - rr * K;
    sA[rr * (K + 4) + cc] = A[(row0 + rr) * K + cc];
  }
  __syncthreads();

  const int lane = tid & 31;
  const int half = lane >> 4;               // 0: lanes 0-15, 1: lanes 16-31
  const int r    = lane & 15;
  const int n0   = (tid >> 5) * 16;         // wave's column tile

  v8f acc = {};
#pragma unroll
  for (int k0 = 0; k0 < K; k0 += 4) {
    const int ka = k0 + 2 * half;
    v2f a, b;
    a.x = sA[r * (K + 4) + ka];
    a.y = sA[r * (K + 4) + ka + 1];
    b.x = W[(size_t)ka * HID + n0 + r];     // 64KB weight matrix: L2-resident broadcast
    b.y = W[(size_t)(ka + 1) * HID + n0 + r];
    acc = __builtin_amdgcn_wmma_f32_16x16x4_f32(false, a, false, b,
                                                (short)0, acc, false, false);
  }

#pragma unroll
  for (int j = 0; j < 8; ++j)
    C[(row0 + j + 8 * half) * HID + n0 + r] = acc[j];
}

// ---------------- message passing ----------------
// acc[i,:] = t[i,:] * dinv[i]^2   (self-loop term; also zero-initializes accumulator)
__global__ void k_selfloop(const float* __restrict__ t, const float* __restrict__ dinv,
                           float* __restrict__ acc, int n) {
  size_t i = (size_t)blockIdx.x * blockDim.x + threadIdx.x;
  if (i < (size_t)n * HID) {
    float dv = dinv[(int)(i >> 7)];
    acc[i] = t[i] * dv * dv;
  }
}

// one wave32 per edge: lane l covers channels l, l+32, l+64, l+96 (coalesced)
__global__ void k_edge_scatter(const int* __restrict__ src, const int* __restrict__ dst,
                               const float* __restrict__ dinv, const float* __restrict__ t,
                               float* __restrict__ acc, int e) {
  int w = (int)(((size_t)blockIdx.x * blockDim.x + threadIdx.x) >> 5);
  int lane = threadIdx.x & 31;
  if (w >= e) return;
  int s = src[w], d = dst[w];
  float norm = dinv[s] * dinv[d];
  const float* ts = t + (size_t)s * HID;
  float* ad = acc + (size_t)d * HID;
#pragma unroll
  for (int j = 0; j < 4; ++j) {
    int k = lane + j * 32;
    atomicAdd(&ad[k], ts[k] * norm);        // global_atomic_add_f32
  }
}

__global__ void k_bias_relu(float* __restrict__ acc, const float* __restrict__ b, int n) {
  size_t i = (size_t)blockIdx.x * blockDim.x + threadIdx.x;
  if (i < (size_t)n * HID) {
    float v = acc[i] + b[(int)(i & (HID - 1))];
    acc[i] = v > 0.f ? v : 0.f;
  }
}

// ---------------- classifier: out[i] = h[i,:] . Wc + bc ----------------
__global__ void k_classify(const float* __restrict__ h, const float* __restrict__ Wc,
                           const float* __restrict__ bc, float* __restrict__ out, int n) {
  int w = (int)(((size_t)blockIdx.x * blockDim.x + threadIdx.x) >> 5);
  int lane = threadIdx.x & 31;
  if (w >= n) return;
  float sum = 0.f;
#pragma unroll
  for (int j = 0; j < 4; ++j) {
    int k = lane + j * 32;
    sum += h[(size_t)w * HID + k] * Wc[k];
  }
#pragma unroll
  for (int off = 16; off > 0; off >>= 1)
    sum += __shfl_xor(sum, off, 32);        // wave32 reduction
  if (lane == 0) out[w] = sum + bc[0];
}

// ---------------- driver ----------------
extern "C" void kernel_launch(void* const* d_in, const int* in_sizes, int n_in,
                              void* d_out, int out_size, void* d_ws, size_t ws_size,
                              hipStream_t stream) {
  const float* x  = (const float*)d_in[0];   // [N,4]
  const int*   ei = (const int*)d_in[1];     // [2,E]
  const float* W1 = (const float*)d_in[2];   // [4,128]
  const float* b1 = (const float*)d_in[3];   // [128]
  const float* W2 = (const float*)d_in[4];   // [128,128]
  const float* b2 = (const float*)d_in[5];   // [128]
  const float* Wc = (const float*)d_in[6];   // [128]
  const float* bc = (const float*)d_in[7];   // [1]
  float* out = (float*)d_out;

  const int n = in_sizes[0] / NFEAT;         // 100000 (multiple of 16)
  const int e = in_sizes[1] / 2;             // 1600000
  const int* src = ei;
  const int* dst = ei + e;

  // workspace: dinv[N] + 3 ping-pong feature buffers [N,128] f32 (~154 MB total)
  char* ws = (char*)d_ws;
  size_t off = 0;
  auto take = [&](size_t bytes) -> float* {
    float* p = (float*)(ws + off);
    off = (off + bytes + 255) & ~(size_t)255;
    return p;
  };
  float* dinv = take((size_t)n * sizeof(float));
  float* bufA = take((size_t)n * HID * sizeof(float));
  float* bufB = take((size_t)n * HID * sizeof(float));
  float* bufC = take((size_t)n * HID * sizeof(float));

  const int nh_blocks   = (int)(((size_t)n * HID + 255) / 256);
  const int n_blocks    = (n + 255) / 256;
  const int e_blocks    = (e + 255) / 256;
  const int edge_blocks = (int)(((size_t)e * 32 + 255) / 256);
  const int cls_blocks  = (int)(((size_t)n * 32 + 255) / 256);
  const int gemm_blocks = n / 16;

  // normalization: deg = 1 + indegree; dinv = deg^-1/2
  k_deg_init <<<n_blocks, 256, 0, stream>>>(dinv, n);
  k_deg_count<<<e_blocks, 256, 0, stream>>>(dst, dinv, e);
  k_dinv     <<<n_blocks, 256, 0, stream>>>(dinv, n);

  // layer 1: x @ W1 -> bufA; propagate -> bufB; +b1, relu
  k_gemm_wmma<NFEAT><<<gemm_blocks, 256, 0, stream>>>(x, W1, bufA);
  k_selfloop    <<<nh_blocks, 256, 0, stream>>>(bufA, dinv, bufB, n);
  k_edge_scatter<<<edge_blocks, 256, 0, stream>>>(src, dst, dinv, bufA, bufB, e);
  k_bias_relu   <<<nh_blocks, 256, 0, stream>>>(bufB, b1, n);

  // layer 2: bufB @ W2 -> bufA; propagate -> bufC; +b2, relu
  k_gemm_wmma<HID><<<gemm_blocks, 256, 0, stream>>>(bufB, W2, bufA);
  k_selfloop    <<<nh_blocks, 256, 0, stream>>>(bufA, dinv, bufC, n);
  k_edge_scatter<<<edge_blocks, 256, 0, stream>>>(src, dst, dinv, bufA, bufC, e);
  k_bias_relu   <<<nh_blocks, 256, 0, stream>>>(bufC, b2, n);

  // layer 3: bufC @ W2 -> bufA; propagate -> bufB; +b2, relu
  k_gemm_wmma<HID><<<gemm_blocks, 256, 0, stream>>>(bufC, W2, bufA);
  k_selfloop    <<<nh_blocks, 256, 0, stream>>>(bufA, dinv, bufB, n);
  k_edge_scatter<<<edge_blocks, 256, 0, stream>>>(src, dst, dinv, bufA, bufB, e);
  k_bias_relu   <<<nh_blocks, 256, 0, stream>>>(bufB, b2, n);

  // classifier
  k_classify<<<cls_blocks, 256, 0, stream>>>(bufB, Wc, bc, out, n);
}